// Attention_32169305047355
// MI455X (gfx1250) — compile-verified
//
#include <hip/hip_runtime.h>

// Problem constants (match reference)
#define BATCH 32
#define PSI   2048
#define EXTRA 64
#define ROWLEN 2112          // PSI + EXTRA
#define HID   128
#define TPB   256            // 8 wave32 per block
#define TILES_PER_BATCH 16   // 2048 rows / 128 rows-per-block
#define ROWS_PER_BLOCK  128  // 8 waves * 16 rows

typedef float v2f __attribute__((ext_vector_type(2)));
typedef float v8f __attribute__((ext_vector_type(8)));

__global__ __launch_bounds__(TPB) void attn_rank1_softmax_kernel(
    const float* __restrict__ params,
    const float* __restrict__ w_f, const float* __restrict__ w_g,
    const float* __restrict__ w_h, const float* __restrict__ w_v,
    const float* __restrict__ gammap,
    float* __restrict__ out)
{
    __shared__ float psiS[PSI];        // 8 KB: this batch's psi row
    __shared__ float zS[8][16];        // per-wave Z results (16 rows each)
    __shared__ float wS[8][16];        // per-wave W results
    __shared__ float redA[TPB];
    __shared__ float redB[TPB];

    const int t    = threadIdx.x;
    const int lane = t & 31;
    const int wave = t >> 5;
    const int blk  = blockIdx.x;
    const int b    = blk / TILES_PER_BATCH;
    const int tile = blk % TILES_PER_BATCH;
    const float* prow = params + (size_t)b * ROWLEN;

    // ---- stage psi[b, 0:2048] into LDS (2 x float4 per thread) ----
    {
        const float4* src = (const float4*)prow;
        float4* dst = (float4*)psiS;
        dst[t]        = src[t];
        dst[t + TPB]  = src[t + TPB];
    }

    // ---- c_fg = dot(w_f, w_g), c_hv = dot(w_h, w_v) (block tree-reduce) ----
    float pf = 0.f, ph = 0.f;
    if (t < HID) { pf = w_f[t] * w_g[t]; ph = w_h[t] * w_v[t]; }
    redA[t] = pf; redB[t] = ph;
    __syncthreads();
    for (int s = TPB / 2; s > 0; s >>= 1) {
        if (t < s) { redA[t] += redA[t + s]; redB[t] += redB[t + s]; }
        __syncthreads();
    }
    const float c_fg = redA[0];
    const float c_hv = redB[0];
    __syncthreads();

    // ---- per-batch min/max of psi (for stable softmax shift) ----
    float mx = -3.402823466e38f, mn = 3.402823466e38f;
    for (int i = t; i < PSI; i += TPB) {
        float v = psiS[i];
        mx = fmaxf(mx, v);
        mn = fminf(mn, v);
    }
    redA[t] = mx; redB[t] = mn;
    __syncthreads();
    for (int s = TPB / 2; s > 0; s >>= 1) {
        if (t < s) {
            redA[t] = fmaxf(redA[t], redA[t + s]);
            redB[t] = fminf(redB[t], redB[t + s]);
        }
        __syncthreads();
    }
    mx = redA[0];
    mn = redB[0];

    // ---- per-lane row setup (each wave owns 16 rows; lane L -> row m = L&15) ----
    const int  m   = lane & 15;
    const bool hi  = lane >= 16;            // lanes 16-31 hold K=2,3 halves
    const int  rowLocal = tile * ROWS_PER_BLOCK + wave * 16 + m;
    const float a   = psiS[rowLocal] * c_fg;
    const float nsh = -((a >= 0.f) ? a * mx : a * mn);   // -max_q(a*x_q)
    const float bc0 = (m == 0) ? 1.0f : 0.0f;            // B column 0 = ones
    const bool  isn1 = (m == 1);                         // B column 1 = psi chunk

    // ---- WMMA-accumulated reduction: D[m,0]=Z_m, D[m,1]=Sum e*x ----
    v8f acc = {0.f, 0.f, 0.f, 0.f, 0.f, 0.f, 0.f, 0.f};
    const float4* psi4 = (const float4*)psiS;
    #pragma unroll 4
    for (int q = 0; q < PSI / 4; ++q) {
        float4 x4 = psi4[q];                 // LDS broadcast (same addr all lanes)
        // K-split across lane halves (matches 32-bit A 16x4 layout)
        float xs0 = hi ? x4.z : x4.x;        // V0: K=0 (lo) / K=2 (hi)
        float xs1 = hi ? x4.w : x4.y;        // V1: K=1 (lo) / K=3 (hi)
        float e0  = __expf(fmaf(a, xs0, nsh));
        float e1  = __expf(fmaf(a, xs1, nsh));
        // B[k, n]: n==0 -> 1.0, n==1 -> psi[q0+k], else 0 (same K-split)
        float bv0 = isn1 ? xs0 : bc0;
        float bv1 = isn1 ? xs1 : bc0;
        v2f av = {e0, e1};
        v2f bv = {bv0, bv1};
        acc = __builtin_amdgcn_wmma_f32_16x16x4_f32(
            /*neg_a=*/false, av, /*neg_b=*/false, bv,
            /*c_mod=*/(short)0, acc, /*reuse_a=*/false, /*reuse_b=*/false);
    }

    // ---- extract D columns 0 (Z) and 1 (W):
    // C/D layout: VGPR r -> lanes0-15: M=r, N=lane; lanes16-31: M=8+r, N=lane-16
    if (lane == 0 || lane == 16) {
        int base = hi ? 8 : 0;
        #pragma unroll
        for (int r = 0; r < 8; ++r) zS[wave][base + r] = acc[r];
    }
    if (lane == 1 || lane == 17) {
        int base = hi ? 8 : 0;
        #pragma unroll
        for (int r = 0; r < 8; ++r) wS[wave][base + r] = acc[r];
    }
    __syncthreads();

    // ---- finalize 128 rows of this block ----
    const float gamma = gammap[0];
    if (t < ROWS_PER_BLOCK) {
        int w  = t >> 4;
        int mm = t & 15;
        int rl = tile * ROWS_PER_BLOCK + t;    // == tile*128 + w*16 + mm
        float Z  = zS[w][mm];
        float Ws = wS[w][mm];
        float sa = (Ws / Z) * c_hv;
        out[(size_t)b * ROWLEN + rl] = fmaf(gamma, sa, psiS[rl]);
    }

    // ---- copy 'rest' (64 floats per batch) once per batch ----
    if (tile == 0 && t >= 128 && t < 128 + EXTRA) {
        int i = t - 128;
        out[(size_t)b * ROWLEN + PSI + i] = prow[PSI + i];
    }
}

extern "C" void kernel_launch(void* const* d_in, const int* in_sizes, int n_in,
                              void* d_out, int out_size, void* d_ws, size_t ws_size,
                              hipStream_t stream) {
    (void)in_sizes; (void)n_in; (void)out_size; (void)d_ws; (void)ws_size;
    const float* params = (const float*)d_in[0];
    const float* w_f    = (const float*)d_in[1];
    const float* w_g    = (const float*)d_in[2];
    const float* w_h    = (const float*)d_in[3];
    const float* w_v    = (const float*)d_in[4];
    const float* gamma  = (const float*)d_in[5];
    float* out = (float*)d_out;

    dim3 grid(BATCH * TILES_PER_BATCH);   // 512 blocks
    dim3 block(TPB);                      // 256 threads = 8 wave32
    attn_rank1_softmax_kernel<<<grid, block, 0, stream>>>(
        params, w_f, w_g, w_h, w_v, gamma, out);
}